// TCNNGrid_901943132811
// MI455X (gfx1250) — compile-verified
//
#include <hip/hip_runtime.h>
#include <hip/hip_bf16.h>

// Instant-NGP hash-grid encoding, forward.
//   inputs: [BATCH, 3] f32 in [0,1)
//   table : [8, 2^19, 2] f32
//   out   : [BATCH, 16] f32  (levels concatenated, 2 feats/level)
//
// L2-gather-bound kernel: 32 MB table is L2-resident on MI455X (192 MB L2);
// input/output streams use non-temporal hints so they don't evict it.

namespace {

constexpr int      kLevels   = 8;
constexpr unsigned kHashmap  = 1u << 19;
constexpr unsigned kHashMask = kHashmap - 1u;
constexpr unsigned kPrimeY   = 2654435761u;
constexpr unsigned kPrimeZ   = 805459861u;

typedef float v4f __attribute__((ext_vector_type(4)));  // native vector: OK for NT builtins

struct alignas(8) F2 { float x, y; };   // -> global_load_b64 gathers

__global__ __launch_bounds__(256) void hashgrid_fwd_kernel(
    const float* __restrict__ in,
    const float* __restrict__ table,
    float* __restrict__ out,
    int batch)
{
    const int i = blockIdx.x * blockDim.x + threadIdx.x;
    if (i >= batch) return;

    // Streamed once: non-temporal loads keep the 24 MB input out of L2.
    const size_t ib = 3u * (size_t)i;
    const float x = __builtin_nontemporal_load(in + ib + 0);
    const float y = __builtin_nontemporal_load(in + ib + 1);
    const float z = __builtin_nontemporal_load(in + ib + 2);

    float o[2 * kLevels];

    #pragma unroll
    for (int l = 0; l < kLevels; ++l) {
        const unsigned res   = 16u << l;                 // 16 .. 2048
        const float    scale = (float)(res - 1u);

        const float px = fmaf(x, scale, 0.5f);
        const float py = fmaf(y, scale, 0.5f);
        const float pz = fmaf(z, scale, 0.5f);
        // px,py,pz >= 0.5, so trunc == floor.
        const unsigned x0 = (unsigned)px;
        const unsigned y0 = (unsigned)py;
        const unsigned z0 = (unsigned)pz;
        const float tx = px - (float)x0;
        const float ty = py - (float)y0;
        const float tz = pz - (float)z0;

        // Corner c: bit0 -> x, bit1 -> y, bit2 -> z  (matches _CORNERS).
        unsigned idx[8];
        const bool dense =
            ((unsigned long long)res * res * res) <= (unsigned long long)kHashmap;
        if (dense) {
            // res is a power of two: x + y*res + z*res^2 via shifts.
            const unsigned s  = 4u + (unsigned)l;
            const unsigned dy = 1u << s;
            const unsigned dz = 1u << (2u * s);
            const unsigned b  = x0 + (y0 << s) + (z0 << (2u * s));
            idx[0] = b;            idx[1] = b + 1u;
            idx[2] = b + dy;       idx[3] = b + dy + 1u;
            idx[4] = b + dz;       idx[5] = b + dz + 1u;
            idx[6] = b + dy + dz;  idx[7] = b + dy + dz + 1u;
        } else {
            const unsigned hx0 = x0;            // prime for dim0 is 1
            const unsigned hx1 = x0 + 1u;
            const unsigned hy0 = y0 * kPrimeY;
            const unsigned hy1 = hy0 + kPrimeY; // (y0+1)*P wraps identically
            const unsigned hz0 = z0 * kPrimeZ;
            const unsigned hz1 = hz0 + kPrimeZ;
            idx[0] = (hx0 ^ hy0 ^ hz0) & kHashMask;
            idx[1] = (hx1 ^ hy0 ^ hz0) & kHashMask;
            idx[2] = (hx0 ^ hy1 ^ hz0) & kHashMask;
            idx[3] = (hx1 ^ hy1 ^ hz0) & kHashMask;
            idx[4] = (hx0 ^ hy0 ^ hz1) & kHashMask;
            idx[5] = (hx1 ^ hy0 ^ hz1) & kHashMask;
            idx[6] = (hx0 ^ hy1 ^ hz1) & kHashMask;
            idx[7] = (hx1 ^ hy1 ^ hz1) & kHashMask;
        }

        // 8 independent 8-byte gathers; default (RT) hint -> stays in L2.
        const F2* __restrict__ tbl = (const F2*)table + (size_t)l * kHashmap;
        F2 f[8];
        #pragma unroll
        for (int c = 0; c < 8; ++c) f[c] = tbl[idx[c]];

        // Trilinear weights, factored.
        const float ux = 1.0f - tx, uy = 1.0f - ty, uz = 1.0f - tz;
        const float w00 = uy * uz, w10 = ty * uz, w01 = uy * tz, w11 = ty * tz;
        const float w[8] = { ux * w00, tx * w00, ux * w10, tx * w10,
                             ux * w01, tx * w01, ux * w11, tx * w11 };

        float r0 = 0.0f, r1 = 0.0f;
        #pragma unroll
        for (int c = 0; c < 8; ++c) {
            r0 = fmaf(w[c], f[c].x, r0);
            r1 = fmaf(w[c], f[c].y, r1);
        }
        o[2 * l + 0] = r0;
        o[2 * l + 1] = r1;
    }

    // 64 contiguous bytes per thread: 4x non-temporal b128 stores, coalesced.
    v4f* __restrict__ ov = (v4f*)(out + (size_t)i * (2u * kLevels));
    #pragma unroll
    for (int q = 0; q < 4; ++q) {
        v4f v = { o[4 * q + 0], o[4 * q + 1], o[4 * q + 2], o[4 * q + 3] };
        __builtin_nontemporal_store(v, ov + q);
    }
}

} // namespace

extern "C" void kernel_launch(void* const* d_in, const int* in_sizes, int n_in,
                              void* d_out, int out_size, void* d_ws, size_t ws_size,
                              hipStream_t stream)
{
    const float* in    = (const float*)d_in[0];   // [BATCH, 3]
    const float* table = (const float*)d_in[1];   // [8, 2^19, 2]
    float*       out   = (float*)d_out;           // [BATCH, 16]

    const int batch = in_sizes[0] / 3;
    const int block = 256;                         // 8 waves (wave32) per block
    const int grid  = (batch + block - 1) / block;

    hashgrid_fwd_kernel<<<grid, block, 0, stream>>>(in, table, out, batch);
}